// SepConv_40218073759942
// MI455X (gfx1250) — compile-verified
//
#include <hip/hip_runtime.h>

// ---------------------------------------------------------------------------
// SepConv double block for MI455X (gfx1250, wave32).
// Pointwise 1x1 conv = 128x128 f32 GEMM per pixel tile via V_WMMA_F32_16X16X4_F32.
// BatchNorm training stats fused via shfl reductions + global f32 atomics.
// ---------------------------------------------------------------------------

typedef __attribute__((ext_vector_type(2))) float v2f;
typedef __attribute__((ext_vector_type(8))) float v8f;

#define BATCH 32
#define CCH   128
#define HH    64
#define WW    64
#define NPIX  (BATCH * HH * WW)      // 131072 pixels
#define TILE_W 16                    // pixels per workgroup (one row segment)
#define HALO_COLS 18                 // TILE_W + 2 halo
#define TCOLS 132                    // 128 + 4 pad: lanes hit distinct LDS banks

// One workgroup = 256 threads = 8 waves. Tile: all 128 channels x 16 pixels
// at fixed (n, h, w0). Wave w computes output channels [16w, 16w+16).
__global__ __launch_bounds__(256) void sepconv_block(
    const float* __restrict__ xin,
    const float* __restrict__ pre_scale,   // BN scale of previous block (mode 1)
    const float* __restrict__ pre_shift,   // BN shift of previous block (mode 1)
    int fuse_pre_bn,
    const float* __restrict__ dww,         // [C,1,3,3] depthwise weights
    const float* __restrict__ pww,         // [O,C] pointwise weights
    float* __restrict__ yout,              // raw pointwise output [N,C,H,W]
    float* __restrict__ gsum,              // [C] per-channel running sum
    float* __restrict__ gsumsq)            // [C] per-channel running sum of squares
{
    __shared__ float sIn[3 * CCH * HALO_COLS];   // relu'd input halo: 27.0 KB
    __shared__ float sDW[CCH * 9];               // depthwise weights:  4.5 KB
    __shared__ float sTt[TILE_W][TCOLS];         // dw output, [pixel][chan]: 8.25 KB

    const int tid  = threadIdx.x;
    const int tile = blockIdx.x;                 // 8192 tiles
    const int w0 = (tile & 3) * TILE_W;          // 4 tiles per row
    const int h  = (tile >> 2) & (HH - 1);
    const int n  = tile >> 8;                    // 256 tiles per image

    // ---- stage depthwise weights into LDS ----
    for (int i = tid; i < CCH * 9; i += 256) sDW[i] = dww[i];

    // ---- stage relu'd (+ prev-BN) input halo: rows h-1..h+1, cols w0-1..w0+16 ----
    for (int i = tid; i < 3 * CCH * HALO_COLS; i += 256) {
        int col = i % HALO_COLS;                 // consecutive tids -> consecutive cols
        int rc  = i / HALO_COLS;
        int r   = rc % 3;
        int c   = rc / 3;
        int hh = h - 1 + r;
        int ww = w0 - 1 + col;
        float v = 0.0f;
        if (hh >= 0 && hh < HH && ww >= 0 && ww < WW) {
            v = xin[(((size_t)n * CCH + c) * HH + hh) * WW + ww];
            if (fuse_pre_bn) v = v * pre_scale[c] + pre_shift[c];
            v = fmaxf(v, 0.0f);                  // relu
        }
        sIn[(r * CCH + c) * HALO_COLS + col] = v;
    }
    __syncthreads();

    // ---- depthwise 3x3 -> sTt[pixel][channel] (transposed for B-operand b64 loads) ----
    for (int i = tid; i < CCH * TILE_W; i += 256) {
        int j = i & (TILE_W - 1);
        int c = i >> 4;
        const float* wv = &sDW[c * 9];
        float acc = 0.0f;
        #pragma unroll
        for (int r = 0; r < 3; ++r) {
            const float* row = &sIn[(r * CCH + c) * HALO_COLS + j];
            acc = fmaf(row[0], wv[r * 3 + 0], acc);
            acc = fmaf(row[1], wv[r * 3 + 1], acc);
            acc = fmaf(row[2], wv[r * 3 + 2], acc);
        }
        sTt[j][c] = acc;
    }
    __syncthreads();

    // ---- pointwise GEMM: D[16 chan x 16 pix] += A[16x4] * B[4x16], K = 128 ----
    const int lane = tid & 31;
    const int wave = tid >> 5;          // 0..7
    const int o0   = wave * 16;         // output-channel base of this wave
    const int nn   = lane & 15;         // A: M row index / B,D: N (pixel) index
    const int del  = lane >> 4;         // K sub-pair selector (ISA 16x4 f32 layout)

    // A lane layout (ISA 7.12.2): lane l holds M = l&15, K = {k0+2*del, k0+2*del+1}
    const float* pA = pww + (size_t)(o0 + nn) * CCH + 2 * del;
    // B assumed mirrored: lane l holds N = l&15, K = {k0+2*del, k0+2*del+1}
    const float* pB = &sTt[nn][2 * del];

    v8f acc = {0.f, 0.f, 0.f, 0.f, 0.f, 0.f, 0.f, 0.f};
    #pragma unroll 8
    for (int k0 = 0; k0 < CCH; k0 += 4) {
        v2f a = *(const v2f*)(pA + k0);     // global_load_b64 (L2-resident weights)
        v2f b = *(const v2f*)(pB + k0);     // ds_load_b64, conflict-free via TCOLS pad
        acc = __builtin_amdgcn_wmma_f32_16x16x4_f32(
            /*neg_a=*/false, a, /*neg_b=*/false, b,
            /*c_mod=*/(short)0, acc, /*reuse_a=*/false, /*reuse_b=*/false);
    }

    // ---- store D: lane -> pixel nn, VGPR v -> channel o0 + v + 8*del ----
    #pragma unroll
    for (int v = 0; v < 8; ++v) {
        int o = o0 + v + 8 * del;
        yout[(((size_t)n * CCH + o) * HH + h) * WW + w0 + nn] = acc[v];
    }

    // ---- fused BN stats: reduce the 16 pixels of each half-wave, then atomics ----
    #pragma unroll
    for (int v = 0; v < 8; ++v) {
        float s = acc[v];
        float q = s * s;
        #pragma unroll
        for (int m = 8; m >= 1; m >>= 1) {
            s += __shfl_xor(s, m, 16);
            q += __shfl_xor(q, m, 16);
        }
        if (nn == 0) {
            int o = o0 + v + 8 * del;
            atomicAdd(&gsum[o],   s);
            atomicAdd(&gsumsq[o], q);
        }
    }
}

// ---- per-channel BN finalize: biased batch stats over (N,H,W) ----
__global__ void sepconv_finalize(const float* __restrict__ gsum,
                                 const float* __restrict__ gsumsq,
                                 const float* __restrict__ gamma,
                                 const float* __restrict__ beta,
                                 float* __restrict__ scale,
                                 float* __restrict__ shift)
{
    int c = threadIdx.x;   // 128 threads
    const float inv = 1.0f / (float)NPIX;
    float mean = gsum[c] * inv;
    float var  = gsumsq[c] * inv - mean * mean;
    float sc   = gamma[c] * rsqrtf(var + 1e-5f);
    scale[c] = sc;
    shift[c] = beta[c] - mean * sc;
}

// ---- final elementwise BN apply (block 2 output) ----
__global__ __launch_bounds__(256) void sepconv_apply(
    const float* __restrict__ y, const float* __restrict__ scale,
    const float* __restrict__ shift, float* __restrict__ out)
{
    size_t i4 = (size_t)blockIdx.x * blockDim.x + threadIdx.x;  // float4 index
    int c = (int)((i4 >> 10) & (CCH - 1));                      // 1024 float4 per (n,c) plane row group
    float4 v = ((const float4*)y)[i4];
    float sc = scale[c], sh = shift[c];
    float4 r;
    r.x = fmaf(v.x, sc, sh);
    r.y = fmaf(v.y, sc, sh);
    r.z = fmaf(v.z, sc, sh);
    r.w = fmaf(v.w, sc, sh);
    ((float4*)out)[i4] = r;
}

extern "C" void kernel_launch(void* const* d_in, const int* in_sizes, int n_in,
                              void* d_out, int out_size, void* d_ws, size_t ws_size,
                              hipStream_t stream) {
    const float* x   = (const float*)d_in[0];
    const float* dw1 = (const float*)d_in[1];
    const float* pw1 = (const float*)d_in[2];
    const float* g1  = (const float*)d_in[3];
    const float* b1  = (const float*)d_in[4];
    const float* dw2 = (const float*)d_in[5];
    const float* pw2 = (const float*)d_in[6];
    const float* g2  = (const float*)d_in[7];
    const float* b2  = (const float*)d_in[8];

    const size_t TENS = (size_t)BATCH * CCH * HH * WW;  // 16777216 floats
    float* ws  = (float*)d_ws;
    float* y1  = ws;                 // raw pointwise output of block 1
    float* y2  = ws + TENS;          // raw pointwise output of block 2
    float* st  = ws + 2 * TENS;      // stats scratch: 1024 floats
    float* sum1 = st,       *sq1 = st + 128, *sc1 = st + 256, *sh1 = st + 384;
    float* sum2 = st + 512, *sq2 = st + 640, *sc2 = st + 768, *sh2 = st + 896;

    // zero the atomic accumulators (memset node is graph-capturable)
    hipMemsetAsync(st, 0, 1024 * sizeof(float), stream);

    const int nTiles = NPIX / TILE_W;        // 8192 workgroups
    sepconv_block<<<nTiles, 256, 0, stream>>>(x,  nullptr, nullptr, 0, dw1, pw1, y1, sum1, sq1);
    sepconv_finalize<<<1, CCH, 0, stream>>>(sum1, sq1, g1, b1, sc1, sh1);
    sepconv_block<<<nTiles, 256, 0, stream>>>(y1, sc1, sh1, 1, dw2, pw2, y2, sum2, sq2);
    sepconv_finalize<<<1, CCH, 0, stream>>>(sum2, sq2, g2, b2, sc2, sh2);
    sepconv_apply<<<(int)(TENS / 4 / 256), 256, 0, stream>>>(y2, sc2, sh2, (float*)d_out);
}